// mrla_module_2430951490290
// MI455X (gfx1250) — compile-verified
//
#include <hip/hip_runtime.h>
#include <math.h>

typedef __attribute__((ext_vector_type(2))) float v2f;
typedef __attribute__((ext_vector_type(8))) float v8f;

#define C_DIM   512
#define HW      1024
#define HEADS   16
#define DPH     32
#define INV_SQRT_DPH 0.17677669529663687f  // 1/sqrt(32)

// ---------------------------------------------------------------------------
// Kernel 1: pool[row] = mean over the 1024 spatial elements of row (b,c).
// One 256-thread block per row; each thread loads one float4 (coalesced
// global_load_b128, 4 KB per block), wave32 shuffle tree + tiny LDS combine.
// ---------------------------------------------------------------------------
__global__ void pool_kernel(const float* __restrict__ xt,
                            float* __restrict__ pool) {
    const int row = blockIdx.x;          // 0 .. 32767  (b*512 + c)
    const int tid = threadIdx.x;         // 0 .. 255
    const float4* src = (const float4*)(xt + (size_t)row * HW);
    float4 v = src[tid];
    float s = (v.x + v.y) + (v.z + v.w);
    #pragma unroll
    for (int off = 16; off > 0; off >>= 1)
        s += __shfl_down(s, off, 32);
    __shared__ float wsum[8];
    if ((tid & 31) == 0) wsum[tid >> 5] = s;
    __syncthreads();
    if (tid == 0) {
        float t = 0.f;
        #pragma unroll
        for (int i = 0; i < 8; ++i) t += wsum[i];
        pool[row] = t * (1.0f / (float)HW);
    }
}

// ---------------------------------------------------------------------------
// Kernel 2: per batch -- ECA conv (k=3, zero pad) over channels, then the 16
// per-head length-32 self dot products via V_WMMA_F32_16X16X4_F32:
//   D = Y * Y^T  (16x32 @ 32x16, K tiled by 4, 8 accumulating WMMAs),
// diagonal D[m][m] = ||y_head m||^2.  One wave does the WMMA (EXEC all-ones).
// A-layout 16x4 f32: lane m /m+16 hold K={0,1}/{2,3};  B 4x16 is symmetric,
// so per-lane A == B == y[(lane&15)*32 + 4*kk + k0 + {0,1}].
// C/D diag: lanes 0-7 vgpr=lane (heads 0-7), lanes 24-31 vgpr=lane-24 (heads 8-15).
// ---------------------------------------------------------------------------
__global__ void attn_kernel(const float* __restrict__ pool,
                            const float* __restrict__ conv_w,
                            float* __restrict__ attn) {
    const int b = blockIdx.x;            // 0..63
    const int c = threadIdx.x;           // 0..511
    __shared__ float sy[C_DIM];

    const float* p = pool + b * C_DIM;
    float pm1 = (c > 0)         ? p[c - 1] : 0.f;
    float p0  = p[c];
    float pp1 = (c < C_DIM - 1) ? p[c + 1] : 0.f;
    sy[c] = conv_w[0] * pm1 + conv_w[1] * p0 + conv_w[2] * pp1;
    __syncthreads();

    if (c < 32) {                        // wave 0 only, full EXEC inside
        const int lane = c;
        const int mn   = lane & 15;
        const int k0   = (lane < 16) ? 0 : 2;
        v8f acc = {};
        #pragma unroll
        for (int kk = 0; kk < 8; ++kk) {
            const int base = mn * DPH + kk * 4 + k0;
            v2f ab;
            ab.x = sy[base];
            ab.y = sy[base + 1];
            acc = __builtin_amdgcn_wmma_f32_16x16x4_f32(
                false, ab, false, ab, (short)0, acc, false, false);
        }
        const bool active = (lane < 8) || (lane >= 24);
        if (active) {
            const int comp = lane & 7;
            float d;
            switch (comp) {
                case 0: d = acc[0]; break;
                case 1: d = acc[1]; break;
                case 2: d = acc[2]; break;
                case 3: d = acc[3]; break;
                case 4: d = acc[4]; break;
                case 5: d = acc[5]; break;
                case 6: d = acc[6]; break;
                default: d = acc[7]; break;
            }
            const int head = (lane < 8) ? lane : (lane - 16);
            const float val = d * INV_SQRT_DPH;
            attn[b * HEADS + head] = 1.0f / (1.0f + __expf(-val));
        }
    }
}

// ---------------------------------------------------------------------------
// Kernel 3: streaming elementwise (HBM-bound pass):
//   out = xt * (wv[c]*attn[b,head]) + lambda[c] * ot_1
// float4 per thread -> b128 loads/stores; xt likely hits L2 (fits in 192MB).
// ---------------------------------------------------------------------------
__global__ void out_kernel(const float* __restrict__ xt,
                           const float* __restrict__ ot1,
                           const float* __restrict__ wv,
                           const float* __restrict__ lam,
                           const float* __restrict__ attn,
                           float* __restrict__ out) {
    const size_t i4 = (size_t)blockIdx.x * blockDim.x + threadIdx.x; // float4 idx
    const int row  = (int)(i4 >> 8);          // 256 float4 per (b,c) row
    const int c    = row & (C_DIM - 1);
    const int b    = row >> 9;
    const int head = c >> 5;
    const float s = wv[c] * attn[b * HEADS + head];
    const float l = lam[c];

    const float4 xv = ((const float4*)xt)[i4];
    const float4 ov = ((const float4*)ot1)[i4];
    float4 r;
    r.x = fmaf(xv.x, s, l * ov.x);
    r.y = fmaf(xv.y, s, l * ov.y);
    r.z = fmaf(xv.z, s, l * ov.z);
    r.w = fmaf(xv.w, s, l * ov.w);
    ((float4*)out)[i4] = r;
}

// ---------------------------------------------------------------------------
extern "C" void kernel_launch(void* const* d_in, const int* in_sizes, int n_in,
                              void* d_out, int out_size, void* d_ws, size_t ws_size,
                              hipStream_t stream) {
    const float* xt     = (const float*)d_in[0];   // 64*512*32*32
    const float* ot1    = (const float*)d_in[1];   // 64*512*32*32
    const float* conv_w = (const float*)d_in[2];   // 3
    const float* wv_w   = (const float*)d_in[3];   // 512
    const float* lam    = (const float*)d_in[4];   // 512
    float* out = (float*)d_out;

    float* pool = (float*)d_ws;                    // 32768 floats
    float* attn = pool + 64 * C_DIM;               // 1024 floats

    const int rows = 64 * C_DIM;                   // 32768
    pool_kernel<<<rows, 256, 0, stream>>>(xt, pool);
    attn_kernel<<<64, C_DIM, 0, stream>>>(pool, conv_w, attn);

    const int n4 = 64 * C_DIM * (HW / 4);          // 8,388,608 float4s
    out_kernel<<<n4 / 256, 256, 0, stream>>>(xt, ot1, wv_w, lam, attn, out);
}